// CSSRClassifier_41721312314294
// MI455X (gfx1250) — compile-verified
//
#include <hip/hip_runtime.h>
#include <math.h>

// ---------------------------------------------------------------------------
// CSSR classifier: 100 class-autoencoders, fused WMMA-f32 chain on gfx1250.
// GEMM engine: V_WMMA_F32_16X16X4_F32 (fp32 matrix cores, wave32).
// Each wave computes TWO 16x16 D-tiles along N sharing one A operand load.
// tanh is a branch-free exp-based expansion (v_exp_f32 + v_rcp_f32), avoiding
// libm tanhf's divergent dual-path lowering between WMMA phases.
// ---------------------------------------------------------------------------

#define K_CLS   100
#define C_DIM   512
#define HID     64
#define LAT     32
#define B_DIM   32
#define HW      196                 // 14*14
#define N_COL   (B_DIM * HW)        // 6272
#define NT      64                  // columns per workgroup tile

typedef __attribute__((ext_vector_type(2))) float v2f;
typedef __attribute__((ext_vector_type(8))) float v8f;

// Branch-free tanh: tanh(x) = sign(x) * (1 - e) / (1 + e), e = exp(-2|x|).
// Safe over the full range (large |x| -> e -> 0 -> +/-1, no overflow path).
// Lowers to v_exp_f32 + v_rcp_f32 + a few VALU ops, no exec divergence.
__device__ __forceinline__ float fast_tanh(float v) {
    const float a = __builtin_fabsf(v);
    const float e = __expf(-2.0f * a);
    const float t = (1.0f - e) * __builtin_amdgcn_rcpf(1.0f + e);
    return __builtin_copysignf(t, v);
}

// Two 16x16 fp32 D-tiles (same M rows, two N column groups) accumulated via a
// shared-A K-loop of V_WMMA_F32_16X16X4_F32.
//  Wrow : pointer to W[mrow][ksel]   (row-major, contiguous in K)
//  BcolX: pointer to B[ksel][colX]   (column fixed per lane, row stride ldb)
//  A layout (ISA 7.12.2, 32-bit A 16x4): lanes 0-15 hold K = kk+0/kk+1,
//  lanes 16-31 hold K = kk+2/kk+3, M = lane&15 in both halves.
__device__ __forceinline__ void wmma_tile2_f32(const float* __restrict__ Wrow,
                                               const float* __restrict__ Bcol0,
                                               const float* __restrict__ Bcol1,
                                               int ldb, int kdim,
                                               v8f& acc0, v8f& acc1) {
#pragma unroll 4
    for (int kk = 0; kk < kdim; kk += 4) {
        v2f a, b0, b1;
        a.x  = Wrow[kk];
        a.y  = Wrow[kk + 1];
        b0.x = Bcol0[(size_t)kk * ldb];
        b0.y = Bcol0[(size_t)(kk + 1) * ldb];
        b1.x = Bcol1[(size_t)kk * ldb];
        b1.y = Bcol1[(size_t)(kk + 1) * ldb];
        acc0 = __builtin_amdgcn_wmma_f32_16x16x4_f32(
            false, a, false, b0, (short)0, acc0, false, false);
        acc1 = __builtin_amdgcn_wmma_f32_16x16x4_f32(
            false, a, false, b1, (short)0, acc1, false, false);
    }
}

// Single-tile variant (used where there is one tile per wave).
__device__ __forceinline__ v8f wmma_tile_f32(const float* __restrict__ Wrow,
                                             const float* __restrict__ Bcol,
                                             int ldb, int kdim) {
    v8f acc = {};
#pragma unroll 4
    for (int kk = 0; kk < kdim; kk += 4) {
        v2f a, b;
        a.x = Wrow[kk];
        a.y = Wrow[kk + 1];
        b.x = Bcol[(size_t)kk * ldb];
        b.y = Bcol[(size_t)(kk + 1) * ldb];
        acc = __builtin_amdgcn_wmma_f32_16x16x4_f32(
            false, a, false, b, (short)0, acc, false, false);
    }
    return acc;
}

// Store a 16x16 f32 D tile (C/D VGPR layout: VGPR j -> M = j (+8 for hi
// lanes), N = lane&15) into an LDS [rows][NT] buffer, with tanh.
__device__ __forceinline__ void store_tile_lds_tanh(float* dst, int mt, int nt,
                                                    int lane, v8f d) {
    const int col = nt * 16 + (lane & 15);
    const int mbase = mt * 16 + ((lane < 16) ? 0 : 8);
#pragma unroll
    for (int j = 0; j < 8; ++j)
        dst[(mbase + j) * NT + col] = fast_tanh(d[j]);
}

__global__ void __launch_bounds__(256)
cssr_ae_chain_kernel(const float* __restrict__ x,
                     const float* __restrict__ W_enc,
                     const float* __restrict__ W_lat,
                     const float* __restrict__ W_delat,
                     const float* __restrict__ W_dec,
                     float* __restrict__ fm_out,
                     float* __restrict__ recon_out,
                     float* __restrict__ logits) {
    __shared__ float hs[HID * NT];   // 16 KB : h  tile [64 x 64]
    __shared__ float fs[LAT * NT];   //  8 KB : fm tile [32 x 64]
    __shared__ float dls[HID * NT];  // 16 KB : d  tile [64 x 64]
    __shared__ float errs[NT];       //  L1 recon error per column

    const int k    = blockIdx.x;          // class index
    const int n0   = blockIdx.y * NT;     // first column of this tile
    const int tid  = threadIdx.x;
    const int lane = tid & 31;
    const int wave = tid >> 5;            // 8 waves (wave32)
    const int ksel = (lane < 16) ? 0 : 2; // A/B K-pair select per lane half
    const int lm   = lane & 15;

    if (tid < NT) errs[tid] = 0.0f;
    __syncthreads();

    // Wave -> (mt, nt-pair) used by phases 1/3/4.
    const int ntp  = (wave & 1) * 2;            // first of two N tiles
    const int col0 = ntp * 16 + lm;             // column of tile 0
    const int col1 = col0 + 16;                 // column of tile 1
    const int na   = n0 + col0, nb = n0 + col1; // global columns
    const int ba = na / HW, pa = na % HW;       // (batch, pixel) per lane
    const int bb = nb / HW, pb = nb % HW;       // tiles may straddle batches

    // ---- Phase 1: h = tanh(W_enc[k] @ X)   [64 x NT], K = 512 -------------
    {
        const int mt = wave >> 1;               // 0..3
        const float* Wrow = W_enc + (size_t)k * HID * C_DIM +
                            (size_t)(mt * 16 + lm) * C_DIM + ksel;
        const float* B0 = x + (size_t)ba * C_DIM * HW + (size_t)ksel * HW + pa;
        const float* B1 = x + (size_t)bb * C_DIM * HW + (size_t)ksel * HW + pb;
        v8f a0 = {}, a1 = {};
        wmma_tile2_f32(Wrow, B0, B1, HW, C_DIM, a0, a1);
        store_tile_lds_tanh(hs, mt, ntp, lane, a0);
        store_tile_lds_tanh(hs, mt, ntp + 1, lane, a1);
    }
    __syncthreads();

    // ---- Phase 2: fm = tanh(W_lat[k] @ h)  [32 x NT], K = 64 --------------
    {
        const int mt = wave >> 2;               // 0..1
        const int nt = wave & 3;                // 0..3
        const int col = nt * 16 + lm;
        const int n = n0 + col;
        const int b = n / HW, p = n % HW;
        const float* Wrow = W_lat + (size_t)k * LAT * HID +
                            (size_t)(mt * 16 + lm) * HID + ksel;
        v8f acc = wmma_tile_f32(Wrow, hs + ksel * NT + col, NT, HID);
        const int mbase = mt * 16 + ((lane < 16) ? 0 : 8);
        float* fmo = fm_out + (size_t)b * K_CLS * LAT * HW +
                     (size_t)k * LAT * HW + p;
#pragma unroll
        for (int j = 0; j < 8; ++j) {
            const float v = fast_tanh(acc[j]);
            fs[(mbase + j) * NT + col] = v;
            fmo[(size_t)(mbase + j) * HW] = v;  // fm_out [B,K,LAT,H,W]
        }
    }
    __syncthreads();

    // ---- Phase 3: d = tanh(W_delat[k] @ fm) [64 x NT], K = 32 -------------
    {
        const int mt = wave >> 1;               // 0..3
        const float* Wrow = W_delat + (size_t)k * HID * LAT +
                            (size_t)(mt * 16 + lm) * LAT + ksel;
        v8f a0 = {}, a1 = {};
        wmma_tile2_f32(Wrow, fs + ksel * NT + col0, fs + ksel * NT + col1,
                       NT, LAT, a0, a1);
        store_tile_lds_tanh(dls, mt, ntp, lane, a0);
        store_tile_lds_tanh(dls, mt, ntp + 1, lane, a1);
    }
    __syncthreads();

    // ---- Phase 4: recon = W_dec[k] @ d  [512 x NT], K = 64; fused L1 err --
    {
        const float* Wbase = W_dec + (size_t)k * C_DIM * HID;
        const float* Bc0 = dls + ksel * NT + col0;
        const float* Bc1 = dls + ksel * NT + col1;
        const float* xc0 = x + (size_t)ba * C_DIM * HW + pa;   // x[ba, :, pa]
        const float* xc1 = x + (size_t)bb * C_DIM * HW + pb;
        float* rc0 = recon_out + (size_t)ba * K_CLS * C_DIM * HW +
                     (size_t)k * C_DIM * HW + pa;
        float* rc1 = recon_out + (size_t)bb * K_CLS * C_DIM * HW +
                     (size_t)k * C_DIM * HW + pb;
        float e0 = 0.0f, e1 = 0.0f;
        for (int ct = (wave >> 1); ct < C_DIM / 16; ct += 4) {
            const float* Wrow = Wbase + (size_t)(ct * 16 + lm) * HID + ksel;
            v8f a0 = {}, a1 = {};
            wmma_tile2_f32(Wrow, Bc0, Bc1, NT, HID, a0, a1);
            const int mbase = ct * 16 + ((lane < 16) ? 0 : 8);
#pragma unroll
            for (int j = 0; j < 8; ++j) {
                const size_t c = (size_t)(mbase + j);
                const float r0 = a0[j], r1 = a1[j];
                rc0[c * HW] = r0;                 // recon_out [B,K,C,H,W]
                rc1[c * HW] = r1;
                e0 += __builtin_fabsf(r0 - xc0[c * HW]);
                e1 += __builtin_fabsf(r1 - xc1[c * HW]);
            }
        }
        atomicAdd(&errs[col0], e0);               // ds_add_f32
        atomicAdd(&errs[col1], e1);
    }
    __syncthreads();

    // logits[k][n] = clamp(-0.1 * sum_c |recon - x|, +/-500)
    if (tid < NT) {
        float e = errs[tid] * (-0.1f);
        e = fminf(fmaxf(e, -500.0f), 500.0f);
        logits[(size_t)k * N_COL + (n0 + tid)] = e;
    }
}

// Per-(b,h,w) softmax over the 100 classes + HxW mean pools into g / error.
__global__ void __launch_bounds__(224)
cssr_softmax_pool_kernel(const float* __restrict__ logits,
                         float* __restrict__ g,
                         float* __restrict__ err_out) {
    __shared__ float g_s[K_CLS];
    __shared__ float e_s[K_CLS];
    const int b   = blockIdx.x;
    const int tid = threadIdx.x;
    if (tid < K_CLS) { g_s[tid] = 0.0f; e_s[tid] = 0.0f; }
    __syncthreads();

    if (tid < HW) {
        const int n = b * HW + tid;
        float mx = -1e30f;
        for (int kk = 0; kk < K_CLS; ++kk)
            mx = fmaxf(mx, logits[(size_t)kk * N_COL + n]);
        float ssum = 0.0f;
        for (int kk = 0; kk < K_CLS; ++kk)
            ssum += __expf(logits[(size_t)kk * N_COL + n] - mx);
        const float inv = 1.0f / ssum;
        const float scale = 1.0f / (float)HW;
        for (int kk = 0; kk < K_CLS; ++kk) {
            const float L = logits[(size_t)kk * N_COL + n];
            const float soft = __expf(L - mx) * inv + 1e-10f;
            atomicAdd(&g_s[kk], soft * scale);
            atomicAdd(&e_s[kk], L * scale);
        }
    }
    __syncthreads();
    if (tid < K_CLS) {
        g[(size_t)b * K_CLS + tid]       = g_s[tid];   // g     [B,K]
        err_out[(size_t)b * K_CLS + tid] = e_s[tid];   // error [B,K]
    }
}

extern "C" void kernel_launch(void* const* d_in, const int* in_sizes, int n_in,
                              void* d_out, int out_size, void* d_ws,
                              size_t ws_size, hipStream_t stream) {
    const float* x       = (const float*)d_in[0];
    const float* W_enc   = (const float*)d_in[1];
    const float* W_lat   = (const float*)d_in[2];
    const float* W_delat = (const float*)d_in[3];
    const float* W_dec   = (const float*)d_in[4];

    // d_out = (g, error, fm_out, recon_out) flat-concatenated (all fp32).
    float* out       = (float*)d_out;
    float* g         = out;                                       // [32,100]
    float* err       = g + (size_t)B_DIM * K_CLS;                 // [32,100]
    float* fm_out    = err + (size_t)B_DIM * K_CLS;               // [B,K,LAT,H,W]
    float* recon_out = fm_out + (size_t)B_DIM * K_CLS * LAT * HW; // [B,K,C,H,W]

    float* logits = (float*)d_ws;   // [K_CLS, N_COL] = 2.5 MB scratch

    dim3 grid(K_CLS, N_COL / NT);   // 100 x 98 workgroups
    cssr_ae_chain_kernel<<<grid, 256, 0, stream>>>(
        x, W_enc, W_lat, W_delat, W_dec, fm_out, recon_out, logits);

    cssr_softmax_pool_kernel<<<B_DIM, 224, 0, stream>>>(logits, g, err);
}